// PEQ_9028021256650
// MI455X (gfx1250) — compile-verified
//
#include <hip/hip_runtime.h>

typedef __attribute__((ext_vector_type(2))) float v2f;
typedef __attribute__((ext_vector_type(8))) float v8f;

// group-of-32 ds_swizzle pattern: lane' = ((lane & 0x1F) | 0) ^ 0x10
#define SWZ_XOR16 0x401F

__device__ __forceinline__ float swz16(float x) {
    return __int_as_float(__builtin_amdgcn_ds_swizzle(__float_as_int(x), SWZ_XOR16));
}

__device__ __forceinline__ void cmul(float& cr, float& ci,
                                     float ar, float ai, float br, float bi) {
    cr = ar * br - ai * bi;
    ci = ar * bi + ai * br;
}

// One wave processes TPW tiles of 16 complex elements.
// WMMA mapping per tile:  D(16 filters x 16 elements) = A(16x4 coeffs) * B(4x16 powers of 1/x)
//   A[m][j] = coeff_j of filter m (real beta or alpha), j=3 column zero
//   B[j][n] = {1, w, w^2, 0}[j] for element n, where w = 1/x  (re / im in two B matrices)
__global__ __launch_bounds__(256) void peq_wmma_kernel(
    const float* __restrict__ xr, const float* __restrict__ xi,
    const float* __restrict__ f, const float* __restrict__ R,
    const float* __restrict__ G,
    float2* __restrict__ out, int N, int tilesPerWave)
{
    const int lane = threadIdx.x & 31;
    const int k    = lane & 15;       // filter index (for A) / element-in-tile (for B)
    const bool lo  = lane < 16;

    // ---------- per-wave setup: biquad coefficients -> A matrices ----------
    float fk = f[k], Rk = R[k], Gk = G[k];
    float mLP = (k == 0) ? Gk : 1.0f;
    float mBP = 2.0f * Rk * ((k < 2) ? sqrtf(Gk) : Gk);
    float mHP = (k == 1) ? Gk : 1.0f;
    float f2  = fk * fk;
    float b0 = f2 * mLP + fk * mBP + mHP;
    float b1 = 2.0f * f2 * mLP - 2.0f * mHP;
    float b2 = f2 * mLP - fk * mBP + mHP;
    float a0 = f2 + 2.0f * Rk * fk + 1.0f;
    float a1 = 2.0f * f2 - 2.0f;
    float a2 = f2 - 2.0f * Rk * fk + 1.0f;

    // A-matrix f32 16x4 layout: lanes 0-15 -> {col0, col1}, lanes 16-31 -> {col2, 0}
    v2f Abeta, Aalpha;
    Abeta[0]  = lo ? b0 : b2;   Abeta[1]  = lo ? b1 : 0.0f;
    Aalpha[0] = lo ? a0 : a2;   Aalpha[1] = lo ? a1 : 0.0f;

    // scalarize the wave index so the tile loop is a scalar branch (EXEC stays all-1)
    const int wave  = __builtin_amdgcn_readfirstlane(
                          (int)((blockIdx.x * blockDim.x + threadIdx.x) >> 5));
    const int base0 = wave * tilesPerWave * 16;

    for (int t = 0; t < tilesPerWave; ++t) {
        int base = base0 + t * 16;
        if (base >= N) break;                     // wave-uniform (scalar)
        int e  = base + k;
        int ec = e < N ? e : N - 1;               // clamp loads for safety (N%16==0 normally)

        // w = 1/x = conj(x)/|x|^2 ; w2 = w*w
        float x_r = xr[ec], x_i = xi[ec];
        float inv = __builtin_amdgcn_rcpf(x_r * x_r + x_i * x_i);
        float wre = x_r * inv, wim = -x_i * inv;
        float w2re = wre * wre - wim * wim;
        float w2im = 2.0f * wre * wim;

        // B-matrix f32 4x16 layout: lanes 0-15 -> {row0, row1}, lanes 16-31 -> {row2, row3=0}
        v2f Bre, Bim;
        Bre[0] = lo ? 1.0f : w2re;   Bre[1] = lo ? wre : 0.0f;
        Bim[0] = lo ? 0.0f : w2im;   Bim[1] = lo ? wim : 0.0f;

        v8f zero = {};
        v8f nre = __builtin_amdgcn_wmma_f32_16x16x4_f32(false, Abeta,  false, Bre, (short)0, zero, false, false);
        v8f nim = __builtin_amdgcn_wmma_f32_16x16x4_f32(false, Abeta,  false, Bim, (short)0, zero, false, false);
        v8f dre = __builtin_amdgcn_wmma_f32_16x16x4_f32(false, Aalpha, false, Bre, (short)0, zero, false, false);
        v8f dim = __builtin_amdgcn_wmma_f32_16x16x4_f32(false, Aalpha, false, Bim, (short)0, zero, false, false);

        // in-lane product over the 8 filter rows this lane holds
        float pnr = nre[0], pni = nim[0];
        float pdr = dre[0], pdi = dim[0];
        #pragma unroll
        for (int r = 1; r < 8; ++r) {
            float tr, ti;
            cmul(tr, ti, pnr, pni, nre[r], nim[r]); pnr = tr; pni = ti;
            cmul(tr, ti, pdr, pdi, dre[r], dim[r]); pdr = tr; pdi = ti;
        }
        // combine with the partner half-wave (filters 8..15 vs 0..7 of same element)
        float onr = swz16(pnr), oni = swz16(pni);
        float odr = swz16(pdr), odi = swz16(pdi);
        float tr, ti;
        cmul(tr, ti, pnr, pni, onr, oni); pnr = tr; pni = ti;
        cmul(tr, ti, pdr, pdi, odr, odi); pdr = tr; pdi = ti;

        // H = (1 + i) * Pnum * conj(Pden) / |Pden|^2
        float dinv = __builtin_amdgcn_rcpf(pdr * pdr + pdi * pdi);
        float hre = (pnr * pdr + pni * pdi) * dinv;
        float him = (pni * pdr - pnr * pdi) * dinv;
        float outr = hre - him;
        float outi = hre + him;

        if (lo && e < N) {
            out[e] = make_float2(outr, outi);   // lanes 0..15: one contiguous 128B burst
        }
    }
}

extern "C" void kernel_launch(void* const* d_in, const int* in_sizes, int n_in,
                              void* d_out, int out_size, void* d_ws, size_t ws_size,
                              hipStream_t stream) {
    const float* xr = (const float*)d_in[0];
    const float* xi = (const float*)d_in[1];
    const float* f  = (const float*)d_in[2];
    const float* R  = (const float*)d_in[3];
    const float* G  = (const float*)d_in[4];
    float2* out = (float2*)d_out;
    const int N = in_sizes[0];

    const int TPW = 16;                       // tiles (of 16 elems) per wave
    const int elemsPerWave  = TPW * 16;       // 256
    const int wavesPerBlock = 8;              // 256 threads, wave32
    long totalWaves = ((long)N + elemsPerWave - 1) / elemsPerWave;
    int blocks = (int)((totalWaves + wavesPerBlock - 1) / wavesPerBlock);

    peq_wmma_kernel<<<blocks, 256, 0, stream>>>(xr, xi, f, R, G, out, N, TPW);
}